// NEUROPULSBonus_unitaryNxN_56358560858274
// MI455X (gfx1250) — compile-verified
//
#include <hip/hip_runtime.h>
#include <math.h>

#define NROW 256
#define NSTG 128     // total mesh stages (M,D,M[,C])
#define NCHUNK 16
#define CSTG 8       // stages per chunk -> half-bandwidth 16 -> block-tridiagonal
#define BAND 34      // band slots per column during W build

typedef float v8f __attribute__((ext_vector_type(8)));
typedef float v2f __attribute__((ext_vector_type(2)));

// ---- physical constants (IL_MMI=0.02, IMB=0.01, IL_CR=0.02, CT=0.01) ----
// a_mmi*t, a_mmi*r, a_cr*sqrt(CT), a_cr*sqrt(1-CT)  (thru == ASQ)
#define AT_C  (sqrtf(0.98f) * sqrtf(0.505f))
#define AR_C  (sqrtf(0.98f) * sqrtf(0.495f))
#define ACT_C (sqrtf(0.98f) * sqrtf(0.01f))
#define ASQ_C (sqrtf(0.98f) * sqrtf(0.99f))

// ------------------------------------------------------------------
// 1) phase table: phR/phI[i*256+g] = exp(i*theta[i][g])
// ------------------------------------------------------------------
__global__ void phase_k(const float* __restrict__ th,
                        float* __restrict__ phR, float* __restrict__ phI) {
  int idx = blockIdx.x * blockDim.x + threadIdx.x;
  if (idx >= 130 * NROW) return;
  float t = th[idx];
  phR[idx] = cosf(t);
  phI[idx] = sinf(t);
}

// ------------------------------------------------------------------
// 2) build chunk operators W_c (block-tridiagonal, 16x16 complex tiles)
//    One thread per (chunk, column): propagate e_j through 8 stages over a
//    34-wide band kept entirely in registers (all inner loops unrolled).
//    Tile storage: Wt[((c*16+rb)*3+s)*2+p][16][16], s: k-block = rb+s-1,
//    p: 0=real 1=imag.
// ------------------------------------------------------------------
__global__ void wbuild_k(const float* __restrict__ phR,
                         const float* __restrict__ phI,
                         float* __restrict__ Wt) {
  int tid = blockIdx.x * blockDim.x + threadIdx.x;
  if (tid >= NCHUNK * NROW) return;
  int c = tid >> 8;          // chunk 0..15
  int j = tid & 255;         // column 0..255
  int base = (j & ~1) - 16;  // even; band rows g = base + t, parity(g)=parity(t)

  const float AT = AT_C, AR = AR_C, ACT = ACT_C, ASQ = ASQ_C;

  float sr[BAND], si[BAND];
#pragma unroll
  for (int t = 0; t < BAND; ++t) { sr[t] = 0.f; si[t] = 0.f; }
  if (j - base == 16) sr[16] = 1.f; else sr[17] = 1.f;

  for (int k = 0; k < CSTG; ++k) {
    int i = c * CSTG + k + 1;          // global stage 1..128
    const float* pR = phR + i * NROW;
    const float* pI = phI + i * NROW;

    // ---- M (even pairs): y0 = AT*x0 + i*AR*x1 ; y1 = i*AR*x0 + AT*x1
#pragma unroll
    for (int t = 0; t < BAND; t += 2) {
      float x0r = sr[t], x0i = si[t], x1r = sr[t+1], x1i = si[t+1];
      sr[t]   = AT * x0r - AR * x1i;
      si[t]   = AT * x0i + AR * x1r;
      sr[t+1] = AT * x1r - AR * x0i;
      si[t+1] = AT * x1i + AR * x0r;
    }
    // ---- D (row phases)
#pragma unroll
    for (int t = 0; t < BAND; ++t) {
      int g = base + t;
      if (g >= 0 && g < NROW) {
        float cs = pR[g], sn = pI[g];
        float xr = sr[t], xi = si[t];
        sr[t] = xr * cs - xi * sn;
        si[t] = xr * sn + xi * cs;
      }
    }
    // ---- M again
#pragma unroll
    for (int t = 0; t < BAND; t += 2) {
      float x0r = sr[t], x0i = si[t], x1r = sr[t+1], x1i = si[t+1];
      sr[t]   = AT * x0r - AR * x1i;
      si[t]   = AT * x0i + AR * x1r;
      sr[t+1] = AT * x1r - AR * x0i;
      si[t+1] = AT * x1i + AR * x0r;
    }
    // ---- C (odd pairs; rows 0 & 255 pass-through), skipped for stage 128
    if (i < NSTG) {
#pragma unroll
      for (int t = 0; t < BAND; ++t) {
        int g = base + t;
        if (g == 0 || g == NROW - 1) { sr[t] *= ASQ; si[t] *= ASQ; }
      }
#pragma unroll
      for (int t = 1; t + 1 < BAND; t += 2) {
        int g = base + t;                       // odd global row
        if (g >= 1 && g <= NROW - 3) {          // pairs (1,2)..(253,254)
          float x0r = sr[t], x0i = si[t], x1r = sr[t+1], x1i = si[t+1];
          sr[t]   = ACT * x0r - ASQ * x1i;
          si[t]   = ACT * x0i + ASQ * x1r;
          sr[t+1] = ACT * x1r - ASQ * x0i;
          si[t+1] = ACT * x1i + ASQ * x0r;
        }
      }
    }
  }

  // fold final phase layer D_129 into the last chunk (left-multiply = row scale)
  if (c == NCHUNK - 1) {
    const float* pR = phR + (NSTG + 1) * NROW;
    const float* pI = phI + (NSTG + 1) * NROW;
#pragma unroll
    for (int t = 0; t < BAND; ++t) {
      int g = base + t;
      if (g >= 0 && g < NROW) {
        float cs = pR[g], sn = pI[g];
        float xr = sr[t], xi = si[t];
        sr[t] = xr * cs - xi * sn;
        si[t] = xr * sn + xi * cs;
      }
    }
  }

  // ---- scatter column j into its 3 row-block tiles (zero then overwrite)
  int kb = j >> 4, col = j & 15;
  for (int drb = -1; drb <= 1; ++drb) {
    int rb = kb + drb;
    if (rb < 0 || rb >= 16) continue;
    int s = kb - rb + 1;
    float* tR = Wt + (size_t)(((c * 16 + rb) * 3 + s) * 2 + 0) * 256;
    float* tI = Wt + (size_t)(((c * 16 + rb) * 3 + s) * 2 + 1) * 256;
    for (int i2 = 0; i2 < 16; ++i2) { tR[i2 * 16 + col] = 0.f; tI[i2 * 16 + col] = 0.f; }
  }
#pragma unroll
  for (int t = 0; t < BAND; ++t) {
    int g = base + t;
    if (g < 0 || g >= NROW) continue;
    int rb = g >> 4;
    int s = kb - rb + 1;
    if (s < 0 || s > 2) continue;
    float* tR = Wt + (size_t)(((c * 16 + rb) * 3 + s) * 2 + 0) * 256;
    float* tI = Wt + (size_t)(((c * 16 + rb) * 3 + s) * 2 + 1) * 256;
    tR[(g & 15) * 16 + col] = sr[t];
    tI[(g & 15) * 16 + col] = si[t];
  }
}

// ------------------------------------------------------------------
// 3) X0 = diag(exp(i*theta[0]))  (split real/imag planes)
// ------------------------------------------------------------------
__global__ void xinit_k(const float* __restrict__ phR, const float* __restrict__ phI,
                        float* __restrict__ XR, float* __restrict__ XI) {
  int idx = blockIdx.x * blockDim.x + threadIdx.x;
  if (idx >= NROW * NROW) return;
  int r = idx >> 8, cc = idx & 255;
  float d = (r == cc) ? 1.f : 0.f;
  XR[idx] = d * phR[cc];
  XI[idx] = d * phI[cc];
}

// ------------------------------------------------------------------
// 4) X <- W_c * X : block-tridiagonal complex matmul on the WMMA pipes.
//    One wave per 16x16 output tile (rb,cb); 3 k-tiles; per k-tile four real
//    16x16x16 products via V_WMMA_F32_16X16X4_F32 (K split into 4 steps).
// ------------------------------------------------------------------
__global__ void applyW_k(const float* __restrict__ Wt, int c,
                         const float* __restrict__ XinR, const float* __restrict__ XinI,
                         float* __restrict__ XoutR, float* __restrict__ XoutI,
                         float* __restrict__ outC /* interleaved complex64 or null */) {
  int wave = (blockIdx.x * blockDim.x + threadIdx.x) >> 5;  // 0..255 (uniform in wave)
  int lane = threadIdx.x & 31;
  int rb = wave >> 4, cb = wave & 15;
  int m = lane & 15;   // A row / B col / D col
  int h = lane >> 4;   // lane half selects K pair (A/B) and M+8 (D)

  v8f accRR = {0.f,0.f,0.f,0.f,0.f,0.f,0.f,0.f};
  v8f accII = {0.f,0.f,0.f,0.f,0.f,0.f,0.f,0.f};
  v8f accI  = {0.f,0.f,0.f,0.f,0.f,0.f,0.f,0.f};

  for (int s = 0; s < 3; ++s) {
    int kb = rb + s - 1;
    if (kb < 0 || kb > 15) continue;  // wave-uniform: EXEC stays all-ones
    const float* tR = Wt + (size_t)(((c * 16 + rb) * 3 + s) * 2 + 0) * 256;
    const float* tI = Wt + (size_t)(((c * 16 + rb) * 3 + s) * 2 + 1) * 256;
    const float* xR = XinR + (size_t)(kb * 16) * NROW + cb * 16;
    const float* xI = XinI + (size_t)(kb * 16) * NROW + cb * 16;
#pragma unroll
    for (int kk = 0; kk < 4; ++kk) {
      int k0 = 4 * kk + 2 * h;
      // A (16x4): lane m holds row m, K = k0, k0+1 (row-major tile, contiguous)
      v2f aR = *(const v2f*)(tR + m * 16 + k0);
      v2f aI = *(const v2f*)(tI + m * 16 + k0);
      // B (4x16): lane m holds col m, K = k0, k0+1
      v2f bR, bI;
      bR.x = xR[(size_t)k0 * NROW + m];
      bR.y = xR[(size_t)(k0 + 1) * NROW + m];
      bI.x = xI[(size_t)k0 * NROW + m];
      bI.y = xI[(size_t)(k0 + 1) * NROW + m];
      // complex product: Yr = Wr*Xr - Wi*Xi ; Yi = Wr*Xi + Wi*Xr
      accRR = __builtin_amdgcn_wmma_f32_16x16x4_f32(false, aR, false, bR, (short)0, accRR, false, false);
      accII = __builtin_amdgcn_wmma_f32_16x16x4_f32(false, aI, false, bI, (short)0, accII, false, false);
      accI  = __builtin_amdgcn_wmma_f32_16x16x4_f32(false, aR, false, bI, (short)0, accI,  false, false);
      accI  = __builtin_amdgcn_wmma_f32_16x16x4_f32(false, aI, false, bR, (short)0, accI,  false, false);
    }
  }

  // D layout: VGPR rr -> row rr + 8*h, col m
#pragma unroll
  for (int rr = 0; rr < 8; ++rr) {
    int g = rb * 16 + rr + 8 * h;
    int ccol = cb * 16 + m;
    float yr = accRR[rr] - accII[rr];
    float yi = accI[rr];
    if (outC) {
      outC[((size_t)g * NROW + ccol) * 2 + 0] = yr;
      outC[((size_t)g * NROW + ccol) * 2 + 1] = yi;
    } else {
      XoutR[(size_t)g * NROW + ccol] = yr;
      XoutI[(size_t)g * NROW + ccol] = yi;
    }
  }
}

// ------------------------------------------------------------------
extern "C" void kernel_launch(void* const* d_in, const int* in_sizes, int n_in,
                              void* d_out, int out_size, void* d_ws, size_t ws_size,
                              hipStream_t stream) {
  (void)in_sizes; (void)n_in; (void)out_size; (void)ws_size;
  const float* thetas = (const float*)d_in[0];   // [130][256] float32

  float* ws  = (float*)d_ws;
  float* phR = ws;                                // 130*256
  float* phI = phR + 130 * NROW;                  // 130*256
  float* Wt  = phI + 130 * NROW;                  // 16*16*3*2*256 = 393216
  float* XAr = Wt + (size_t)NCHUNK * 16 * 3 * 2 * 256;
  float* XAi = XAr + NROW * NROW;
  float* XBr = XAi + NROW * NROW;
  float* XBi = XBr + NROW * NROW;                 // total ~2.9 MB

  phase_k<<<(130 * NROW + 255) / 256, 256, 0, stream>>>(thetas, phR, phI);
  wbuild_k<<<(NCHUNK * NROW + 255) / 256, 256, 0, stream>>>(phR, phI, Wt);
  xinit_k<<<(NROW * NROW + 255) / 256, 256, 0, stream>>>(phR, phI, XAr, XAi);

  float *inR = XAr, *inI = XAi, *outR = XBr, *outI = XBi;
  for (int c = 0; c < NCHUNK; ++c) {
    bool last = (c == NCHUNK - 1);
    applyW_k<<<32, 256, 0, stream>>>(Wt, c, inR, inI,
                                     last ? nullptr : outR,
                                     last ? nullptr : outI,
                                     last ? (float*)d_out : nullptr);
    float* t;
    t = inR; inR = outR; outR = t;
    t = inI; inI = outI; outI = t;
  }
}